// GraphLearner_62740882260818
// MI455X (gfx1250) — compile-verified
//
#include <hip/hip_runtime.h>
#include <math.h>

typedef __attribute__((ext_vector_type(2))) float v2f;
typedef __attribute__((ext_vector_type(8))) float v8f;

#define WB 8
#define WN 1024
#define WD 128
#define WP 16
#define EPSF 1e-5f

static __device__ __forceinline__ float warpSum32(float v) {
    v += __shfl_xor(v, 16, 32);
    v += __shfl_xor(v, 8, 32);
    v += __shfl_xor(v, 4, 32);
    v += __shfl_xor(v, 2, 32);
    v += __shfl_xor(v, 1, 32);
    return v;
}

// ---------------------------------------------------------------------------
// Kernel 1: V[b,n,p*D+d] = context[b,n,d]*w[p,d] / max(||context[b,n,:]*w[p,:]||, 1e-12)
// One wave per (b,n,p); 128 d-elements -> float4 per lane, wave32 reduction.
// ---------------------------------------------------------------------------
__global__ __launch_bounds__(256) void k_prep_v(const float* __restrict__ ctx,
                                                const float* __restrict__ w,
                                                float* __restrict__ V) {
    int wid = threadIdx.x >> 5, lane = threadIdx.x & 31;
    int gid = blockIdx.x * 8 + wid;          // 0 .. B*N*P-1
    int p = gid & (WP - 1);
    int n = (gid >> 4) & (WN - 1);
    int b = gid >> 14;
    const float4* c4 = (const float4*)(ctx + ((size_t)b * WN + n) * WD);
    const float4* w4 = (const float4*)(w + (size_t)p * WD);
    float4 cv = c4[lane];
    float4 wv = w4[lane];
    float4 pv = make_float4(cv.x * wv.x, cv.y * wv.y, cv.z * wv.z, cv.w * wv.w);
    float ss = pv.x * pv.x + pv.y * pv.y + pv.z * pv.z + pv.w * pv.w;
    ss = warpSum32(ss);
    float inv = 1.0f / fmaxf(sqrtf(ss), 1e-12f);
    float4 ov = make_float4(pv.x * inv, pv.y * inv, pv.z * inv, pv.w * inv);
    float4* vout = (float4*)(V + (((size_t)b * WN + n) * WP + p) * WD);
    vout[lane] = ov;
}

// ---------------------------------------------------------------------------
// Kernel 2: temp_acc[b,n] = sqrt(au^2+av^2+eps) * |cos(uv_angleACC - uv_angle)|
// ---------------------------------------------------------------------------
__global__ void k_tacc(const float* __restrict__ au, const float* __restrict__ av,
                       const float* __restrict__ ua, const float* __restrict__ uaacc,
                       float* __restrict__ tacc) {
    int i = blockIdx.x * blockDim.x + threadIdx.x;
    if (i >= WB * WN) return;
    float A = au[i] * au[i] + av[i] * av[i];
    tacc[i] = sqrtf(A + EPSF) * fabsf(cosf(uaacc[i] - ua[i]));
}

// ---------------------------------------------------------------------------
// Kernel 3: batched SYRK  C[b] = post(scale * X[b] X[b]^T + addc)
// X: (B, N=1024, K) row-major.  Wave tile 64x32 via V_WMMA_F32_16X16X4_F32.
// Per lane: one b128 load supplies K=8 (two wmma k-steps); A/B fragments use
// an identical K-bijection so products pair correctly.
// ---------------------------------------------------------------------------
template <bool RELU>
__global__ __launch_bounds__(256) void k_syrk(const float* __restrict__ X,
                                              float* __restrict__ C,
                                              int K, float scale, float addc) {
    const int N = WN;
    int lane = threadIdx.x & 31;
    int wv = threadIdx.x >> 5;
    int wr = wv >> 2;                 // 0..1  -> 64 rows each
    int wc = wv & 3;                  // 0..3  -> 32 cols each
    int b = blockIdx.z;
    const float* Xb = X + (size_t)b * N * K;
    float* Cb = C + (size_t)b * N * N;
    int rBase = blockIdx.x * 128 + wr * 64;
    int cBase = blockIdx.y * 128 + wc * 32;
    int lr = lane & 15;               // row/col within 16-tile
    int kl = (lane >> 4) * 4;         // lanes 0-15: k0..k0+3, lanes 16-31: k0+4..k0+7

    const float* arow[4];
    const float* brow[2];
#pragma unroll
    for (int i = 0; i < 4; i++) arow[i] = Xb + (size_t)(rBase + 16 * i + lr) * K + kl;
#pragma unroll
    for (int j = 0; j < 2; j++) brow[j] = Xb + (size_t)(cBase + 16 * j + lr) * K + kl;

    v8f acc[4][2] = {};
    for (int k0 = 0; k0 < K; k0 += 8) {
        float4 a[4], bb[2];
#pragma unroll
        for (int i = 0; i < 4; i++) a[i] = *(const float4*)(arow[i] + k0);
#pragma unroll
        for (int j = 0; j < 2; j++) bb[j] = *(const float4*)(brow[j] + k0);
#pragma unroll
        for (int i = 0; i < 4; i++)
#pragma unroll
            for (int j = 0; j < 2; j++) {
                v2f a0 = {a[i].x, a[i].y};
                v2f b0 = {bb[j].x, bb[j].y};
                acc[i][j] = __builtin_amdgcn_wmma_f32_16x16x4_f32(
                    false, a0, false, b0, (short)0, acc[i][j], false, false);
                v2f a1 = {a[i].z, a[i].w};
                v2f b1 = {bb[j].z, bb[j].w};
                acc[i][j] = __builtin_amdgcn_wmma_f32_16x16x4_f32(
                    false, a1, false, b1, (short)0, acc[i][j], false, false);
            }
    }

    // C/D layout: VGPR v -> row v (lanes 0-15) or v+8 (lanes 16-31), col = lane&15
    int rofs = (lane & 16) ? 8 : 0;
#pragma unroll
    for (int i = 0; i < 4; i++)
#pragma unroll
        for (int j = 0; j < 2; j++) {
            int col = cBase + 16 * j + lr;
#pragma unroll
            for (int v = 0; v < 8; v++) {
                int row = rBase + 16 * i + rofs + v;
                float x = acc[i][j][v] * scale + addc;
                if (RELU) x = fmaxf(x, 0.0f);
                Cb[(size_t)row * N + col] = x;
            }
        }
}

// ---------------------------------------------------------------------------
// Kernel 4: selfwind row (b,n): |spd| * |cos(angle[n,m]-uv[b,n]) + sigma*diag| / (dist+eps)
// then L2-normalize the row.  One 256-thread block per row, 4 elems/thread.
// ---------------------------------------------------------------------------
__global__ __launch_bounds__(256) void k_selfwind(const float* __restrict__ speed,
                                                  const float* __restrict__ uv_angle,
                                                  const float* __restrict__ angle,
                                                  const float* __restrict__ dist,
                                                  const float* __restrict__ tacc,
                                                  float* __restrict__ SW) {
    int n = blockIdx.x, b = blockIdx.y;
    float spd = fabsf(speed[b * WN + n]);
    float uva = uv_angle[b * WN + n];
    float ta = tacc[b * WN + n];
    float vals[4];
    float ss = 0.0f;
#pragma unroll
    for (int q = 0; q < 4; q++) {
        int m = threadIdx.x + q * 256;
        float an = angle[(size_t)n * WN + m];
        float dd = dist[(size_t)n * WN + m] + EPSF;
        float c = cosf(an - uva);
        if (m == n) c += 0.5f * ta;          // sigma = 0.5, diagonal acc_part
        float v = spd * fabsf(c) / dd;
        vals[q] = v;
        ss += v * v;
    }
    __shared__ float red[8];
    float s = warpSum32(ss);
    if ((threadIdx.x & 31) == 0) red[threadIdx.x >> 5] = s;
    __syncthreads();
    float tot = 0.0f;
#pragma unroll
    for (int i = 0; i < 8; i++) tot += red[i];
    float inv = 1.0f / fmaxf(sqrtf(tot), 1e-12f);
    float* out = SW + ((size_t)b * WN + n) * WN;
#pragma unroll
    for (int q = 0; q < 4; q++) out[threadIdx.x + q * 256] = vals[q] * inv;
}

// ---------------------------------------------------------------------------
// Kernel 5: per row: y = l2norm(y_raw); t = |theta*sw + sqrt(y)| + eps;
// u = l2norm(t); out = attention * u
// ---------------------------------------------------------------------------
__global__ __launch_bounds__(256) void k_final(const float* __restrict__ ATT,
                                               const float* __restrict__ SW,
                                               const float* __restrict__ Y,
                                               float* __restrict__ OUT) {
    int n = blockIdx.x, b = blockIdx.y;
    size_t base = ((size_t)b * WN + n) * WN;
    int lane = threadIdx.x & 31, wid = threadIdx.x >> 5;
    float yv[4], sv[4];
    float ssy = 0.0f;
#pragma unroll
    for (int q = 0; q < 4; q++) {
        int m = threadIdx.x + q * 256;
        yv[q] = Y[base + m];
        sv[q] = SW[base + m];
        ssy += yv[q] * yv[q];
    }
    __shared__ float red[8];
    float s = warpSum32(ssy);
    if (lane == 0) red[wid] = s;
    __syncthreads();
    float toty = 0.0f;
#pragma unroll
    for (int i = 0; i < 8; i++) toty += red[i];
    float invy = 1.0f / fmaxf(sqrtf(toty), 1e-12f);

    float tv[4];
    float sst = 0.0f;
#pragma unroll
    for (int q = 0; q < 4; q++) {
        float t = fabsf(0.5f * sv[q] + sqrtf(yv[q] * invy)) + EPSF;  // theta = 0.5
        tv[q] = t;
        sst += t * t;
    }
    __syncthreads();                    // red re-use
    s = warpSum32(sst);
    if (lane == 0) red[wid] = s;
    __syncthreads();
    float tott = 0.0f;
#pragma unroll
    for (int i = 0; i < 8; i++) tott += red[i];
    float invt = 1.0f / fmaxf(sqrtf(tott), 1e-12f);
#pragma unroll
    for (int q = 0; q < 4; q++) {
        int m = threadIdx.x + q * 256;
        OUT[base + m] = ATT[base + m] * tv[q] * invt;
    }
}

// ---------------------------------------------------------------------------
extern "C" void kernel_launch(void* const* d_in, const int* in_sizes, int n_in,
                              void* d_out, int out_size, void* d_ws, size_t ws_size,
                              hipStream_t stream) {
    const float* context     = (const float*)d_in[0];
    const float* acc_u       = (const float*)d_in[1];
    const float* acc_v       = (const float*)d_in[2];
    const float* uv_angle    = (const float*)d_in[3];
    const float* uv_angleACC = (const float*)d_in[4];
    const float* speed       = (const float*)d_in[5];
    /* d_in[6] = isPhy (unused by reference) */
    const float* dist        = (const float*)d_in[7];
    const float* angle       = (const float*)d_in[8];
    const float* wt          = (const float*)d_in[9];

    char* ws = (char*)d_ws;
    float* V    = (float*)(ws);                   // 64 MiB  (B*N*P*D f32)
    float* Y    = (float*)(ws);                   // 32 MiB, reuses V region after attention
    float* ATT  = (float*)(ws + (64ull << 20));   // 32 MiB
    float* SW   = (float*)(ws + (96ull << 20));   // 32 MiB
    float* TACC = (float*)(ws + (128ull << 20));  // 32 KiB

    k_prep_v<<<(WB * WN * WP) / 8, 256, 0, stream>>>(context, wt, V);
    k_tacc<<<(WB * WN + 255) / 256, 256, 0, stream>>>(acc_u, acc_v, uv_angle, uv_angleACC, TACC);

    dim3 g(WN / 128, WN / 128, WB);
    // attention = ReLU( V V^T / P )
    k_syrk<true><<<g, 256, 0, stream>>>(V, ATT, WP * WD, 1.0f / (float)WP, 0.0f);

    k_selfwind<<<dim3(WN, WB), 256, 0, stream>>>(speed, uv_angle, angle, dist, TACC, SW);

    // y_raw = (gamma/20) * SW SW^T + eps
    k_syrk<false><<<g, 256, 0, stream>>>(SW, Y, WN, 1.4f / 20.0f, EPSF);

    k_final<<<dim3(WN, WB), 256, 0, stream>>>(ATT, SW, Y, (float*)d_out);
}